// GATLayer_5420248728143
// MI455X (gfx1250) — compile-verified
//
#include <hip/hip_runtime.h>
#include <hip/hip_bf16.h>
#include <math.h>

typedef __attribute__((ext_vector_type(2))) float v2f;
typedef __attribute__((ext_vector_type(8))) float v8f;

#define F_IN   128
#define HEADS  4
#define OUTH   32
#define FOUT   128   // HEADS * OUTH
#define NEG_SLOPE 0.2f

// ---- order-preserving float <-> uint map for atomicMax-based segment max ----
__device__ __forceinline__ unsigned ordkey(float f) {
    unsigned u = __float_as_uint(f);
    return u ^ ((u >> 31) ? 0xFFFFFFFFu : 0x80000000u);
}
__device__ __forceinline__ float keyord(unsigned k) {
    unsigned u = (k >> 31) ? (k ^ 0x80000000u) : ~k;
    return __uint_as_float(u);
}

// ---- init: zero out & denom, init segmax keys, build Wcat[f][k*32+h] ----
__global__ void gat_init(float* __restrict__ out, float* __restrict__ denom,
                         unsigned* __restrict__ segmax, float* __restrict__ wcat,
                         const float* __restrict__ W, int N) {
    int i = blockIdx.x * blockDim.x + threadIdx.x;
    if (i < N * FOUT) out[i] = 0.0f;
    if (i < HEADS * N) { denom[i] = 0.0f; segmax[i] = ordkey(-1.0e9f); }
    if (i < F_IN * FOUT) {
        int f = i / FOUT, c = i % FOUT;
        int k = c / OUTH, h = c % OUTH;
        wcat[i] = W[(k * F_IN + f) * OUTH + h];   // W: [HEADS][F_IN][OUTH]
    }
}

// ---- Wh = x @ Wcat via V_WMMA_F32_16X16X4_F32 (exact fp32 matmul) ----
// block = 256 threads (8 waves). Block b computes rows [16b,16b+16); wave w
// computes cols [16w,16w+16). K loop: 128/4 = 32 WMMA issues per wave.
__global__ void __launch_bounds__(256) gat_gemm_wmma(const float* __restrict__ x,
                                                     const float* __restrict__ wcat,
                                                     float* __restrict__ wh, int N) {
    const int lane = threadIdx.x & 31;
    const int wave = threadIdx.x >> 5;
    int row0 = blockIdx.x * 16;
    if (row0 + 16 > N) row0 = N - 16;      // uniform clamp, EXEC stays all-ones
    const int col0 = wave * 16;
    const int m    = lane & 15;            // M (A) / N (B,D) coordinate
    const int half = lane >> 4;            // K-half selector

    const float* __restrict__ arow = x + (size_t)(row0 + m) * F_IN;
    v8f acc = {};
    #pragma unroll 4
    for (int k0 = 0; k0 < F_IN; k0 += 4) {
        const int ka = k0 + half * 2;
        v2f a, b;
        a.x = arow[ka];                               // A[m][ka]
        a.y = arow[ka + 1];                           // A[m][ka+1]
        b.x = wcat[(size_t)ka       * FOUT + col0 + m]; // B[ka][n]
        b.y = wcat[(size_t)(ka + 1) * FOUT + col0 + m]; // B[ka+1][n]
        acc = __builtin_amdgcn_wmma_f32_16x16x4_f32(
            /*neg_a=*/false, a, /*neg_b=*/false, b,
            /*c_mod=*/(short)0, acc, /*reuse_a=*/false, /*reuse_b=*/false);
    }
    // D layout: VGPR v, half -> M = v + 8*half ; lane&15 -> N
    float* __restrict__ drow = wh + (size_t)row0 * FOUT + col0 + m;
    #pragma unroll
    for (int v = 0; v < 8; ++v)
        drow[(size_t)(v + half * 8) * FOUT] = acc[v];
}

// ---- e_l[k][n], e_r[k][n] = <Wh[n, k*32:+32], a_l/a_r[k]> ----
__global__ void gat_coef(const float* __restrict__ wh,
                         const float* __restrict__ a_l, const float* __restrict__ a_r,
                         float* __restrict__ el, float* __restrict__ er, int N) {
    int i = blockIdx.x * blockDim.x + threadIdx.x;   // i = k*N + n
    if (i >= HEADS * N) return;
    int k = i / N, n = i % N;
    const float* __restrict__ row = wh + (size_t)n * FOUT + k * OUTH;
    float sl = 0.0f, sr = 0.0f;
    #pragma unroll
    for (int h = 0; h < OUTH; ++h) {
        float v = row[h];
        sl = fmaf(v, a_l[k * OUTH + h], sl);
        sr = fmaf(v, a_r[k * OUTH + h], sr);
    }
    el[i] = sl; er[i] = sr;
}

// ---- per-edge logits + segment max (ordered-uint atomicMax) ----
__global__ void gat_logits(const int* __restrict__ src, const int* __restrict__ dst,
                           const float* __restrict__ el, const float* __restrict__ er,
                           float* __restrict__ ebuf, unsigned* __restrict__ segmax,
                           int N, int E) {
    int e = blockIdx.x * blockDim.x + threadIdx.x;
    if (e >= E) return;
    int s = src[e], d = dst[e];
    #pragma unroll
    for (int k = 0; k < HEADS; ++k) {
        float v = el[k * N + s] + er[k * N + d];
        v = (v >= 0.0f) ? v : NEG_SLOPE * v;         // leaky_relu
        ebuf[(size_t)k * E + e] = v;
        atomicMax(&segmax[k * N + d], ordkey(v));
    }
}

// ---- exp(e - segmax[dst]) ; denom[dst] += exp ----
__global__ void gat_exp(const int* __restrict__ dst, float* __restrict__ ebuf,
                        const unsigned* __restrict__ segmax, float* __restrict__ denom,
                        int N, int E) {
    int e = blockIdx.x * blockDim.x + threadIdx.x;
    if (e >= E) return;
    int d = dst[e];
    #pragma unroll
    for (int k = 0; k < HEADS; ++k) {
        float m  = keyord(segmax[k * N + d]);
        float ex = expf(ebuf[(size_t)k * E + e] - m);
        ebuf[(size_t)k * E + e] = ex;
        atomicAdd(&denom[k * N + d], ex);
    }
}

// ---- aggregation: out[dst, :] += alpha * Wh[src, :]  (1 wave per edge) ----
__global__ void __launch_bounds__(256) gat_aggregate(const int* __restrict__ src,
                                                     const int* __restrict__ dst,
                                                     const float* __restrict__ ebuf,
                                                     const float* __restrict__ denom,
                                                     const float* __restrict__ wh,
                                                     float* __restrict__ out,
                                                     int N, int E) {
    int tid  = blockIdx.x * blockDim.x + threadIdx.x;
    int e    = tid >> 5;
    int lane = tid & 31;
    if (e >= E) return;
    int s = src[e], d = dst[e];
    float alpha[HEADS];
    #pragma unroll
    for (int k = 0; k < HEADS; ++k)
        alpha[k] = ebuf[(size_t)k * E + e] / (denom[k * N + d] + 1e-16f);
    const float* __restrict__ srow = wh + (size_t)s * FOUT;
    float* __restrict__ drow = out + (size_t)d * FOUT;
    #pragma unroll
    for (int k = 0; k < HEADS; ++k) {
        int c = k * OUTH + lane;                     // lane = feature within head
        atomicAdd(&drow[c], alpha[k] * srow[c]);
    }
}

extern "C" void kernel_launch(void* const* d_in, const int* in_sizes, int n_in,
                              void* d_out, int out_size, void* d_ws, size_t ws_size,
                              hipStream_t stream) {
    const float* x   = (const float*)d_in[0];
    const int*   ei  = (const int*)d_in[1];     // edge_index flattened [2, E]
    const float* W   = (const float*)d_in[2];   // [HEADS, F_IN, OUTH]
    const float* a_l = (const float*)d_in[3];
    const float* a_r = (const float*)d_in[4];

    const int N = in_sizes[0] / F_IN;
    const int E = in_sizes[1] / 2;
    const int* src = ei;
    const int* dst = ei + E;

    // workspace carve-out (256B aligned slices)
    char*  ws  = (char*)d_ws;
    size_t off = 0;
    auto carve = [&](size_t bytes) -> void* {
        void* p = ws + off;
        off = (off + bytes + 255) & ~(size_t)255;
        return p;
    };
    float*    wcat   = (float*)   carve((size_t)F_IN * FOUT * sizeof(float));
    float*    wh     = (float*)   carve((size_t)N * FOUT * sizeof(float));
    float*    el     = (float*)   carve((size_t)HEADS * N * sizeof(float));
    float*    er     = (float*)   carve((size_t)HEADS * N * sizeof(float));
    unsigned* segmax = (unsigned*)carve((size_t)HEADS * N * sizeof(unsigned));
    float*    denom  = (float*)   carve((size_t)HEADS * N * sizeof(float));
    float*    ebuf   = (float*)   carve((size_t)HEADS * E * sizeof(float));
    (void)ws_size; (void)n_in; (void)out_size;

    float* out = (float*)d_out;

    // 1) init (covers N*FOUT elements, rounded up to whole 256-thread blocks)
    {
        long total = (long)N * FOUT;
        int blocks = (int)((total + 255) / 256);
        gat_init<<<blocks, 256, 0, stream>>>(out, denom, segmax, wcat, W, N);
    }
    // 2) WMMA GEMM: 16-row stripes, 8 waves cover all 128 output cols
    {
        int blocks = (N + 15) / 16;
        gat_gemm_wmma<<<blocks, 256, 0, stream>>>(x, wcat, wh, N);
    }
    // 3) attention coefficients
    {
        int total = HEADS * N;
        gat_coef<<<(total + 255) / 256, 256, 0, stream>>>(wh, a_l, a_r, el, er, N);
    }
    // 4) edge logits + segment max
    gat_logits<<<(E + 255) / 256, 256, 0, stream>>>(src, dst, el, er, ebuf, segmax, N, E);
    // 5) exp + segment sum
    gat_exp<<<(E + 255) / 256, 256, 0, stream>>>(dst, ebuf, segmax, denom, N, E);
    // 6) wave-per-edge weighted aggregation
    {
        long threads = (long)E * 32;
        int blocks = (int)((threads + 255) / 256);
        gat_aggregate<<<blocks, 256, 0, stream>>>(src, dst, ebuf, denom, wh, out, N, E);
    }
}